// GIN_38225208934940
// MI455X (gfx1250) — compile-verified
//
#include <hip/hip_runtime.h>
#include <hip/hip_bf16.h>
#include <math.h>

// ---------------------------------------------------------------------------
// GIN forward for MI455X (gfx1250, wave32).
//   h = relu(x @ W_in + b_in)
//   3x: agg = h + scatter_add(h[src] -> dst);  h = relu(agg @ W_mlp + b)
//   out = log_softmax(h @ W_out + b_out)
// GEMMs use V_WMMA_F32_16X16X4_F32 (exact fp32 matrix math).
// ---------------------------------------------------------------------------

typedef __attribute__((ext_vector_type(2))) float v2f;
typedef __attribute__((ext_vector_type(8))) float v8f;

#define NHID 128

// ---------------------------------------------------------------------------
// Y[r, 0:128] = relu(X[r, 0:K] @ W[K,128] + bias[128])
// Block: 256 threads (8 waves). Block tile: 16 rows x 128 cols.
// Wave w computes the 16x16 tile at columns [16w, 16w+16).
// LDS pad +4: row stride K+4 floats -> 16B-aligned rows (b128 LDS ops) and
// bank-conflict-free A-fragment reads (stride mod 64 banks = 4; 16 lanes ->
// 16 distinct banks).
// ---------------------------------------------------------------------------
template <int K>
__global__ __launch_bounds__(256) void linear_relu_wmma(
    const float* __restrict__ X, const float* __restrict__ W,
    const float* __restrict__ bias, float* __restrict__ Y, int n_rows) {
  constexpr int KP = K + 4;
  constexpr int K4 = K / 4;
  __shared__ float Xs[16 * KP];

  const int rowBase = blockIdx.x * 16;

  // Cooperative float4 load of the 16xK input tile (coalesced along K).
  for (int idx = threadIdx.x; idx < 16 * K4; idx += 256) {
    const int r = idx / K4;               // power of two -> shift
    const int c4 = idx - r * K4;
    const int gr = rowBase + r;
    float4 v = make_float4(0.f, 0.f, 0.f, 0.f);
    if (gr < n_rows) v = ((const float4*)(X + (size_t)gr * K))[c4];
    *(float4*)&Xs[r * KP + c4 * 4] = v;   // 16B aligned: KP*4 % 16 == 0
  }
  __syncthreads();

  const int lane = threadIdx.x & 31;
  const int wave = threadIdx.x >> 5;      // 0..7
  const int colBase = wave << 4;          // 16 columns per wave
  const int m = lane & 15;
  const int kh = (lane >> 4) << 1;        // lanes 0-15 -> K offset 0, lanes 16-31 -> 2

  v8f acc = {};
#pragma unroll 8
  for (int k = 0; k < K; k += 4) {
    // A fragment (16x4 f32): lane m = row, VGPR0/1 = K = k+kh, k+kh+1
    v2f a, b;
    a.x = Xs[m * KP + k + kh];
    a.y = Xs[m * KP + k + kh + 1];
    // B fragment (4x16 f32): VGPR j holds rows {j, j+2}; cols striped on lanes
    const float* wp = W + (size_t)(k + kh) * NHID + colBase + m;
    b.x = wp[0];
    b.y = wp[NHID];
    acc = __builtin_amdgcn_wmma_f32_16x16x4_f32(
        /*neg_a=*/false, a, /*neg_b=*/false, b,
        /*c_mod=*/(short)0, acc, /*reuse_a=*/false, /*reuse_b=*/false);
  }

  // C/D layout: VGPR j -> row j (lanes 0-15) / row j+8 (lanes 16-31), col = lane&15
  const int rOff = (lane < 16) ? 0 : 8;
  const float bc = bias[colBase + m];
  float* yp = Y + (size_t)(rowBase + rOff) * NHID + colBase + m;

  if (rowBase + 16 <= n_rows) {           // full tile: straight-line stores
#pragma unroll
    for (int j = 0; j < 8; ++j) {
      yp[(size_t)j * NHID] = fmaxf(acc[j] + bc, 0.0f);
    }
  } else {                                // tail tile (not hit for N=50000)
#pragma unroll
    for (int j = 0; j < 8; ++j) {
      if (rowBase + rOff + j < n_rows) {
        yp[(size_t)j * NHID] = fmaxf(acc[j] + bc, 0.0f);
      }
    }
  }
}

// ---------------------------------------------------------------------------
// dst[i] = src[i]  (float4 vectorized; used to seed agg = h before scatter)
// ---------------------------------------------------------------------------
__global__ __launch_bounds__(256) void copy_f4(const float4* __restrict__ src,
                                               float4* __restrict__ dst, int n4) {
  const int i = blockIdx.x * 256 + threadIdx.x;
  if (i < n4) dst[i] = src[i];
}

// ---------------------------------------------------------------------------
// agg[dst[e], :] += h[src[e], :]   one wave per edge, float4 per lane.
// 512B coalesced read per edge; 4 global f32 atomics per lane. Edge indices
// are wave-uniform -> SGPR loads.
// ---------------------------------------------------------------------------
__global__ __launch_bounds__(256) void scatter_add_edges(
    const float* __restrict__ h, const int* __restrict__ esrc,
    const int* __restrict__ edst, float* __restrict__ agg, int n_edges) {
  const int e = blockIdx.x * 8 + (threadIdx.x >> 5);
  if (e >= n_edges) return;
  const int lane = threadIdx.x & 31;
  const int s = esrc[e];
  const int d = edst[e];
  const float4 v = ((const float4*)(h + (size_t)s * NHID))[lane];
  float* a = agg + (size_t)d * NHID + lane * 4;
  atomicAdd(a + 0, v.x);
  atomicAdd(a + 1, v.y);
  atomicAdd(a + 2, v.z);
  atomicAdd(a + 3, v.w);
}

// ---------------------------------------------------------------------------
// out[row, c] = log_softmax(h[row,:] @ W_out[:,c] + b_out[c]), 40 classes.
// One 64-thread block per row; LDS tree reductions for max and sum(exp).
// ---------------------------------------------------------------------------
__global__ __launch_bounds__(64) void out_logsoftmax(
    const float* __restrict__ h, const float* __restrict__ Wout,
    const float* __restrict__ bout, float* __restrict__ out, int n_rows,
    int n_class) {
  const int row = blockIdx.x;
  if (row >= n_rows) return;
  const int t = threadIdx.x;
  const float* hr = h + (size_t)row * NHID;

  float logit = -INFINITY;
  if (t < n_class) {
    float acc = bout[t];
#pragma unroll 8
    for (int k = 0; k < NHID; ++k) acc = fmaf(hr[k], Wout[k * n_class + t], acc);
    logit = acc;
  }

  __shared__ float red[64];
  red[t] = logit;
  __syncthreads();
#pragma unroll
  for (int s = 32; s > 0; s >>= 1) {
    if (t < s) red[t] = fmaxf(red[t], red[t + s]);
    __syncthreads();
  }
  const float mx = red[0];
  __syncthreads();

  red[t] = (t < n_class) ? expf(logit - mx) : 0.0f;
  __syncthreads();
#pragma unroll
  for (int s = 32; s > 0; s >>= 1) {
    if (t < s) red[t] += red[t + s];
    __syncthreads();
  }
  const float lse = logf(red[0]) + mx;

  if (t < n_class) out[(size_t)row * n_class + t] = logit - lse;
}

// ---------------------------------------------------------------------------
extern "C" void kernel_launch(void* const* d_in, const int* in_sizes, int n_in,
                              void* d_out, int out_size, void* d_ws,
                              size_t ws_size, hipStream_t stream) {
  const float* x      = (const float*)d_in[0];  // [N, 256]
  const int*   esrc   = (const int*)d_in[1];    // [E]
  const int*   edst   = (const int*)d_in[2];    // [E]
  const float* W_in   = (const float*)d_in[3];  // [256, 128]
  const float* b_in   = (const float*)d_in[4];  // [128]
  const float* W_mlps = (const float*)d_in[5];  // [3, 128, 128]
  const float* b_mlps = (const float*)d_in[6];  // [3, 128]
  const float* W_out  = (const float*)d_in[7];  // [128, 40]
  const float* b_out  = (const float*)d_in[8];  // [40]
  float* out = (float*)d_out;

  const int n_nodes = in_sizes[0] / 256;
  const int n_edges = in_sizes[1];
  const int n_class = in_sizes[8];
  const int n_layers = in_sizes[6] / NHID;

  float* hbuf = (float*)d_ws;                         // [N, 128]
  float* agg  = hbuf + (size_t)n_nodes * NHID;        // [N, 128]

  const int gemmBlocks = (n_nodes + 15) / 16;
  const int n4 = n_nodes * NHID / 4;
  const int copyBlocks = (n4 + 255) / 256;
  const int scatBlocks = (n_edges + 7) / 8;

  // h = relu(x @ W_in + b_in)
  linear_relu_wmma<256><<<gemmBlocks, 256, 0, stream>>>(x, W_in, b_in, hbuf,
                                                        n_nodes);

  for (int i = 0; i < n_layers; ++i) {
    // agg = h  (GIN eps=0 self term), then agg[d] += h[s] over edges
    copy_f4<<<copyBlocks, 256, 0, stream>>>((const float4*)hbuf, (float4*)agg,
                                            n4);
    scatter_add_edges<<<scatBlocks, 256, 0, stream>>>(hbuf, esrc, edst, agg,
                                                      n_edges);
    // h = relu(agg @ W_mlps[i] + b_mlps[i])
    linear_relu_wmma<128><<<gemmBlocks, 256, 0, stream>>>(
        agg, W_mlps + (size_t)i * NHID * NHID, b_mlps + (size_t)i * NHID, hbuf,
        n_nodes);
  }

  // out = log_softmax(h @ W_out + b_out)
  out_logsoftmax<<<n_nodes, 64, 0, stream>>>(hbuf, W_out, b_out, out, n_nodes,
                                             n_class);
}